// GCNLayer_37469294691137
// MI455X (gfx1250) — compile-verified
//
#include <hip/hip_runtime.h>

#define N_NODES 100000
#define N_EDGES 1600000
#define IN_DIM  128
#define OUT_DIM 64
#define NEG_SLOPE 0.01f

typedef float v2f __attribute__((ext_vector_type(2)));
typedef float v8f __attribute__((ext_vector_type(8)));

// ---------------- degree accumulation (float atomics) ----------------
__global__ void degree_kernel(const int* __restrict__ src, const int* __restrict__ dst,
                              float* __restrict__ outdeg, float* __restrict__ indeg, int E) {
    int e = blockIdx.x * blockDim.x + threadIdx.x;
    if (e < E) {
        atomicAdd(&outdeg[src[e]], 1.0f);
        atomicAdd(&indeg[dst[e]], 1.0f);
    }
}

// ---------------- norms: deg -> rsqrt(max(deg,1)) in place ----------------
__global__ void norm_kernel(float* __restrict__ ns, float* __restrict__ nd, int n) {
    int i = blockIdx.x * blockDim.x + threadIdx.x;
    if (i < n) {
        ns[i] = rsqrtf(fmaxf(ns[i], 1.0f));
        nd[i] = rsqrtf(fmaxf(nd[i], 1.0f));
    }
}

// ---------------- x = (h @ W) * norm_src[row]   via V_WMMA_F32_16X16X4_F32 ----------------
// One wave per 16-row m-tile; 4 n-tiles (OUT=64) accumulated simultaneously.
// A (16x4 f32):  lane l -> m = l&15, VGPR0/1 hold K = kb, kb+1 with kb = 2*(l>>4)
// B (4x16 f32):  lane l -> n = l&15, VGPR0/1 hold K = kb, kb+1
// C/D (16x16):   VGPR r -> row = r + 8*(l>>4), col = l&15
__global__ void __launch_bounds__(256)
gemm_wmma_kernel(const float* __restrict__ h, const float* __restrict__ W,
                 const float* __restrict__ ns, float* __restrict__ x, int mtiles) {
    // W transposed into LDS: Wt[c*130 + k] = W[k*64 + c]  (stride 130 keeps 8B align)
    __shared__ float Wt[OUT_DIM * 130];
    for (int i = threadIdx.x; i < IN_DIM * OUT_DIM; i += 256) {
        int k = i >> 6;
        int c = i & 63;
        Wt[c * 130 + k] = W[i];
    }
    __syncthreads();

    const int wave = threadIdx.x >> 5;
    const int lane = threadIdx.x & 31;
    const int mtile = blockIdx.x * 8 + wave;
    if (mtile >= mtiles) return;   // wave-uniform: EXEC stays all-1s for WMMA

    const int mrow = (mtile << 4) + (lane & 15);
    const int kb   = (lane >> 4) << 1;   // 0 or 2
    const int n    = lane & 15;

    const float* hrow = h + (size_t)mrow * IN_DIM + kb;
    const float* w0 = &Wt[(0 * 16 + n) * 130 + kb];
    const float* w1 = &Wt[(1 * 16 + n) * 130 + kb];
    const float* w2 = &Wt[(2 * 16 + n) * 130 + kb];
    const float* w3 = &Wt[(3 * 16 + n) * 130 + kb];

    v8f acc0 = {}, acc1 = {}, acc2 = {}, acc3 = {};
#pragma unroll 4
    for (int k = 0; k < IN_DIM; k += 4) {
        v2f a  = *(const v2f*)(hrow + k);
        v2f b0 = *(const v2f*)(w0 + k);
        v2f b1 = *(const v2f*)(w1 + k);
        v2f b2 = *(const v2f*)(w2 + k);
        v2f b3 = *(const v2f*)(w3 + k);
        acc0 = __builtin_amdgcn_wmma_f32_16x16x4_f32(false, a, false, b0, (short)0, acc0, false, false);
        acc1 = __builtin_amdgcn_wmma_f32_16x16x4_f32(false, a, false, b1, (short)0, acc1, false, false);
        acc2 = __builtin_amdgcn_wmma_f32_16x16x4_f32(false, a, false, b2, (short)0, acc2, false, false);
        acc3 = __builtin_amdgcn_wmma_f32_16x16x4_f32(false, a, false, b3, (short)0, acc3, false, false);
    }

    const int rowbase = (mtile << 4) + ((lane >> 4) << 3);
#pragma unroll
    for (int r = 0; r < 8; ++r) {
        int row = rowbase + r;
        float s = ns[row];
        float* xr = x + (size_t)row * OUT_DIM + n;
        xr[0]  = acc0[r] * s;
        xr[16] = acc1[r] * s;
        xr[32] = acc2[r] * s;
        xr[48] = acc3[r] * s;
    }
}

// ---------------- agg[dst] += x[src]  (edge gather + scatter-add) ----------------
// One thread per (edge, 4 contiguous columns): coalesced float4 gather, 4 f32 atomics.
__global__ void scatter_kernel(const float* __restrict__ x, const int* __restrict__ src,
                               const int* __restrict__ dst, float* __restrict__ agg, int E) {
    int t = blockIdx.x * blockDim.x + threadIdx.x;   // E*16 threads
    int e  = t >> 4;
    int c4 = (t & 15) << 2;
    if (e < E) {
        int s = src[e];
        int d = dst[e];
        const float4 v = *(const float4*)(x + (size_t)s * OUT_DIM + c4);
        float* ap = agg + (size_t)d * OUT_DIM + c4;
        atomicAdd(ap + 0, v.x);
        atomicAdd(ap + 1, v.y);
        atomicAdd(ap + 2, v.z);
        atomicAdd(ap + 3, v.w);
    }
}

// ---------------- out = LeakyReLU(agg * norm_dst + b) ----------------
__global__ void finalize_kernel(const float* __restrict__ agg, const float* __restrict__ nd,
                                const float* __restrict__ b, float* __restrict__ out, int total) {
    int t = blockIdx.x * blockDim.x + threadIdx.x;
    if (t < total) {
        int i = t >> 6;
        int c = t & 63;
        float v = agg[t] * nd[i] + b[c];
        out[t] = (v >= 0.0f) ? v : v * NEG_SLOPE;
    }
}

extern "C" void kernel_launch(void* const* d_in, const int* in_sizes, int n_in,
                              void* d_out, int out_size, void* d_ws, size_t ws_size,
                              hipStream_t stream) {
    const float* h    = (const float*)d_in[0];
    const float* W    = (const float*)d_in[1];
    const float* b    = (const float*)d_in[2];
    const int*   esrc = (const int*)d_in[3];
    const int*   edst = (const int*)d_in[4];
    float*       out  = (float*)d_out;

    // workspace layout (floats): ns[N] | nd[N] | x[N*64] | agg[N*64]
    float* ws  = (float*)d_ws;
    float* ns  = ws;
    float* nd  = ws + N_NODES;
    float* x   = ws + 2 * (size_t)N_NODES;
    float* agg = x + (size_t)N_NODES * OUT_DIM;

    hipMemsetAsync(ns,  0, 2 * (size_t)N_NODES * sizeof(float), stream);
    hipMemsetAsync(agg, 0, (size_t)N_NODES * OUT_DIM * sizeof(float), stream);

    degree_kernel<<<(N_EDGES + 255) / 256, 256, 0, stream>>>(esrc, edst, ns, nd, N_EDGES);
    norm_kernel<<<(N_NODES + 255) / 256, 256, 0, stream>>>(ns, nd, N_NODES);

    const int mtiles = N_NODES / 16;   // 6250 exactly
    gemm_wmma_kernel<<<(mtiles + 7) / 8, 256, 0, stream>>>(h, W, ns, x, mtiles);

    const long long sthreads = (long long)N_EDGES * 16;
    scatter_kernel<<<(unsigned)((sthreads + 255) / 256), 256, 0, stream>>>(x, esrc, edst, agg, N_EDGES);

    const int total = N_NODES * OUT_DIM;
    finalize_kernel<<<(total + 255) / 256, 256, 0, stream>>>(agg, nd, b, out, total);
}